// HeadUpdator_12249246728580
// MI455X (gfx1250) — compile-verified
//
#include <hip/hip_runtime.h>
#include <math.h>

typedef __attribute__((ext_vector_type(2))) float v2f;
typedef __attribute__((ext_vector_type(4))) float f4;
typedef __attribute__((ext_vector_type(8))) float v8f;

#define HW_TOT 65536.0f

__device__ __forceinline__ float sigmoidf_(float x) {
  return 1.0f / (1.0f + __expf(-x));
}

// ---------------------------------------------------------------------------
// Kernel 1: per-(b,c) plane reductions over feat (256x256 f32 planes).
// Writes 25 floats per plane at ws + bc*32:
//  [0]      = T (total sum)
//  [1..10]  = row sums: rows 0..4 then rows 251..255
//  [11..20] = col sums: cols 0..4 then cols 251..255
//  [21..24] = corners (0,0) (0,255) (255,0) (255,255)
// ---------------------------------------------------------------------------
__global__ void __launch_bounds__(256) feat_reduce(const float* __restrict__ feat,
                                                   float* __restrict__ wsF) {
  int bc = blockIdx.x;                                   // b*64 + c, 1024 blocks
  const f4* plane = (const f4*)(feat + (size_t)bc * 65536);
  __shared__ float sT;
  __shared__ float sRow[10];
  __shared__ float sCol[10];
  __shared__ float sCor[4];
  int t = threadIdx.x;
  if (t == 0) sT = 0.f;
  if (t < 10) { sRow[t] = 0.f; sCol[t] = 0.f; }
  __syncthreads();

  int lane  = t & 63;     // float4 index within a row (coalesced across lanes)
  int rbase = t >> 6;     // 0..3 rows in flight
  float tsum = 0.f;
  float c0=0,c1=0,c2=0,c3=0,c4=0,c5=0,c6=0,c7=0,c8=0,c9=0;

  for (int it = 0; it < 64; ++it) {
    int r = it * 4 + rbase;
    f4 v = __builtin_nontemporal_load(&plane[r * 64 + lane]);  // single-use stream
    float s4 = v.x + v.y + v.z + v.w;
    tsum += s4;
    if (r < 5)        atomicAdd(&sRow[r], s4);
    else if (r > 250) atomicAdd(&sRow[r - 246], s4);   // 251..255 -> 5..9
    if (lane == 0) {
      c0 += v.x; c1 += v.y; c2 += v.z; c3 += v.w;      // cols 0..3
      if (r == 0)   sCor[0] = v.x;
      if (r == 255) sCor[2] = v.x;
    } else if (lane == 1) {
      c4 += v.x;                                       // col 4
    } else if (lane == 62) {
      c5 += v.w;                                       // col 251
    } else if (lane == 63) {
      c6 += v.x; c7 += v.y; c8 += v.z; c9 += v.w;      // cols 252..255
      if (r == 0)   sCor[1] = v.w;
      if (r == 255) sCor[3] = v.w;
    }
  }
  atomicAdd(&sT, tsum);
  if (lane == 0) {
    atomicAdd(&sCol[0], c0); atomicAdd(&sCol[1], c1);
    atomicAdd(&sCol[2], c2); atomicAdd(&sCol[3], c3);
  } else if (lane == 1) {
    atomicAdd(&sCol[4], c4);
  } else if (lane == 62) {
    atomicAdd(&sCol[5], c5);
  } else if (lane == 63) {
    atomicAdd(&sCol[6], c6); atomicAdd(&sCol[7], c7);
    atomicAdd(&sCol[8], c8); atomicAdd(&sCol[9], c9);
  }
  __syncthreads();
  if (t < 25) {
    float v = (t == 0) ? sT
            : (t < 11) ? sRow[t - 1]
            : (t < 21) ? sCol[t - 11]
                       : sCor[t - 21];
    wsF[bc * 32 + t] = v;
  }
}

// ---------------------------------------------------------------------------
// Kernel 2: pred branch. Bilinear 2x upsample (jax half-pixel + boundary
// renormalization => out[0]=in[0], out[2k]=.25 in[k-1]+.75 in[k],
// out[2k+1]=.75 in[k]+.25 in[k+1], out[255]=in[127]), then
// s = sigmoid(up); pa = s + s*(1 - sigmoid(s)).
// Writes per batch at wsP + b*16:
//  [0]=sum(s) [1]=T(pa) [2]=row0(pa) [3]=row255(pa) [4]=col0(pa) [5]=col255(pa)
//  [6..9]=pa corners
// ---------------------------------------------------------------------------
__device__ __forceinline__ void bil_coeff(int o, int n, int* i0, int* i1,
                                          float* w0, float* w1) {
  int k = o >> 1;
  if ((o & 1) == 0) {
    if (k == 0) { *i0 = 0; *i1 = 0; *w0 = 0.f; *w1 = 1.f; }
    else        { *i0 = k - 1; *i1 = k; *w0 = 0.25f; *w1 = 0.75f; }
  } else {
    if (k == n - 1) { *i0 = n - 1; *i1 = n - 1; *w0 = 1.f; *w1 = 0.f; }
    else            { *i0 = k; *i1 = k + 1; *w0 = 0.75f; *w1 = 0.25f; }
  }
}

__global__ void __launch_bounds__(256) pred_reduce(const float* __restrict__ pred,
                                                   float* __restrict__ wsP) {
  int b = blockIdx.x;                       // 16 blocks
  const float* in = pred + (size_t)b * 16384;   // 128x128
  __shared__ float sSumS, sTpa, sRowT, sRowB, sColL, sColR, sCor[4];
  int t = threadIdx.x;
  if (t == 0) { sSumS = 0.f; sTpa = 0.f; sColL = 0.f; sColR = 0.f; }
  __syncthreads();

  int y = t;                                 // one output row per thread
  int iy0, iy1; float wy0, wy1;
  bil_coeff(y, 128, &iy0, &iy1, &wy0, &wy1);
  const float* r0 = in + iy0 * 128;
  const float* r1 = in + iy1 * 128;

  float rs = 0.f, rpa = 0.f, cl = 0.f, cr = 0.f;
  for (int x = 0; x < 256; ++x) {
    int ix0, ix1; float wx0, wx1;
    bil_coeff(x, 128, &ix0, &ix1, &wx0, &wx1);
    float up = wy0 * (wx0 * r0[ix0] + wx1 * r0[ix1]) +
               wy1 * (wx0 * r1[ix0] + wx1 * r1[ix1]);
    float s  = sigmoidf_(up);
    float pa = s * (1.f - sigmoidf_(s)) + s;   // double sigmoid per reference
    rs += s; rpa += pa;
    if (x == 0)   cl = pa;
    if (x == 255) cr = pa;
    if (y == 0)   { if (x == 0) sCor[0] = pa; if (x == 255) sCor[1] = pa; }
    if (y == 255) { if (x == 0) sCor[2] = pa; if (x == 255) sCor[3] = pa; }
  }
  atomicAdd(&sSumS, rs);
  atomicAdd(&sTpa, rpa);
  atomicAdd(&sColL, cl);
  atomicAdd(&sColR, cr);
  if (y == 0)   sRowT = rpa;
  if (y == 255) sRowB = rpa;
  __syncthreads();
  if (t == 0) {
    float* o = wsP + b * 16;
    o[0] = sSumS; o[1] = sTpa; o[2] = sRowT; o[3] = sRowB;
    o[4] = sColL; o[5] = sColR;
    o[6] = sCor[0]; o[7] = sCor[1]; o[8] = sCor[2]; o[9] = sCor[3];
  }
}

// ---------------------------------------------------------------------------
// WMMA f32 16x16xK GEMM: Out(16xN) = A(16xK) @ W^T + bias, W is NxK row-major.
// Executed by one fully-active wave (EXEC all ones). Fragment layout per ISA:
//  A 16x4: lanes 0-15 -> M=lane, K=k0+{0,1}; lanes 16-31 -> K=k0+{2,3}
//  B 4x16: lanes 0-15 -> N=lane, K=k0+{0,1}; lanes 16-31 -> K=k0+{2,3}
//  C/D: VGPR v, lane l -> M = v + 8*(l>=16), N = l%16
// ---------------------------------------------------------------------------
__device__ __forceinline__ void wmma_gemm(const float* A, int lda,
                                          const float* __restrict__ W,
                                          const float* __restrict__ bias,
                                          float* Out, int ldo, int K, int N) {
  int lane = threadIdx.x & 31;
  int half = lane >> 4;       // 0 or 1
  int mn   = lane & 15;
  for (int nt = 0; nt < N; nt += 16) {
    v8f acc = {};
    for (int k0 = 0; k0 < K; k0 += 4) {
      int kb = k0 + half * 2;
      v2f a; a[0] = A[mn * lda + kb];             a[1] = A[mn * lda + kb + 1];
      v2f b; b[0] = W[(size_t)(nt + mn) * K + kb]; b[1] = W[(size_t)(nt + mn) * K + kb + 1];
      acc = __builtin_amdgcn_wmma_f32_16x16x4_f32(
          /*neg_a=*/false, a, /*neg_b=*/false, b,
          /*c_mod=*/(short)0, acc, /*reuse_a=*/false, /*reuse_b=*/false);
    }
    float bv = bias[nt + mn];
    #pragma unroll
    for (int v = 0; v < 8; ++v)
      Out[(v + 8 * half) * ldo + nt + mn] = acc[v] + bv;
  }
}

__device__ __forceinline__ void ln_stats(const float* p, float* mu, float* rs) {
  float m = 0.f;
  for (int i = 0; i < 64; ++i) m += p[i];
  m *= (1.f / 64.f);
  float v = 0.f;
  for (int i = 0; i < 64; ++i) { float d = p[i] - m; v += d * d; }
  v *= (1.f / 64.f);
  *mu = m;
  *rs = 1.0f / sqrtf(v + 1e-5f);
}

struct Params {
  const float *wsF, *wsP, *head;
  const float *dw_w, *dw_b;
  const float *hw_w, *hw_b;   // inc_hw 3x3
  const float *ww_w, *ww_b;   // inc_w  1x11
  const float *hh_w, *hh_b;   // inc_h  11x1
  const float *pt_w, *pt_b, *ht_w, *ht_b;
  const float *pg_w, *pg_b, *hg_w, *hg_b, *fc_w, *fc_b;
  const float *pni_g, *pni_b, *hni_g, *hni_b;
  const float *pno_g, *pno_b, *hno_g, *hno_b, *fcn_g, *fcn_b;
  float* out;
};

// ---------------------------------------------------------------------------
// Kernel 3: finalize conv-sums, form assemble = pred_sum[b]*feat_sum[b,c],
// then the entire head path with WMMA GEMMs + LayerNorms in LDS.
// ---------------------------------------------------------------------------
__global__ void __launch_bounds__(256) head_finalize(Params P) {
  __shared__ float sAsm[16 * 64];
  __shared__ float sHead[16 * 64];
  __shared__ float sPF[16 * 128];
  __shared__ float sHF[16 * 128];
  __shared__ float sGate[16 * 64];
  __shared__ float sHG[16 * 64];
  __shared__ float sPG[16 * 64];
  __shared__ float sUpd[16 * 64];
  __shared__ float sU2[16 * 64];
  __shared__ float sPsum[16];
  __shared__ float sMu[80], sRs[80];
  int t = threadIdx.x;

  // ---- A: feat channel sums (conv boundary correction) ----
  for (int bc = t; bc < 1024; bc += 256) {
    const float* w = P.wsF + bc * 32;
    int c = bc & 63;
    float T = w[0];
    float fs;
    if (c < 40) {
      fs = T;
    } else if (c < 48) {                       // 3x3, pad (1,1)
      int g = c - 40;
      fs = P.hw_b[g] * HW_TOT;
      for (int ky = 0; ky < 3; ++ky) {
        int sy = ky - 1;
        float Rex = (sy > 0) ? w[1] : (sy < 0 ? w[10] : 0.f);   // row0 / row255
        for (int kx = 0; kx < 3; ++kx) {
          int sx = kx - 1;
          float Cex = (sx > 0) ? w[11] : (sx < 0 ? w[20] : 0.f); // col0 / col255
          float cor = 0.f;
          if      (sy ==  1 && sx ==  1) cor = w[21];
          else if (sy ==  1 && sx == -1) cor = w[22];
          else if (sy == -1 && sx ==  1) cor = w[23];
          else if (sy == -1 && sx == -1) cor = w[24];
          fs += P.hw_w[g * 9 + ky * 3 + kx] * (T - Rex - Cex + cor);
        }
      }
    } else if (c < 56) {                       // 1x11, pad (0,5)
      int g = c - 48;
      fs = P.ww_b[g] * HW_TOT;
      for (int kx = 0; kx < 11; ++kx) {
        int sx = kx - 5;
        float Cex = 0.f;
        if (sx > 0)      for (int j = 0; j <  sx; ++j) Cex += w[11 + j];
        else if (sx < 0) for (int j = 0; j < -sx; ++j) Cex += w[20 - j];
        fs += P.ww_w[g * 11 + kx] * (T - Cex);
      }
    } else {                                   // 11x1, pad (5,0)
      int g = c - 56;
      fs = P.hh_b[g] * HW_TOT;
      for (int ky = 0; ky < 11; ++ky) {
        int sy = ky - 5;
        float Rex = 0.f;
        if (sy > 0)      for (int j = 0; j <  sy; ++j) Rex += w[1 + j];
        else if (sy < 0) for (int j = 0; j < -sy; ++j) Rex += w[10 - j];
        fs += P.hh_w[g * 11 + ky] * (T - Rex);
      }
    }
    sAsm[bc] = fs;
  }
  // ---- pred batch sums ----
  if (t < 16) {
    const float* p = P.wsP + t * 16;
    float cs = P.dw_b[0] * HW_TOT;
    for (int ky = 0; ky < 3; ++ky) {
      int sy = ky - 1;
      float Rex = (sy > 0) ? p[2] : (sy < 0 ? p[3] : 0.f);
      for (int kx = 0; kx < 3; ++kx) {
        int sx = kx - 1;
        float Cex = (sx > 0) ? p[4] : (sx < 0 ? p[5] : 0.f);
        float cor = 0.f;
        if      (sy ==  1 && sx ==  1) cor = p[6];
        else if (sy ==  1 && sx == -1) cor = p[7];
        else if (sy == -1 && sx ==  1) cor = p[8];
        else if (sy == -1 && sx == -1) cor = p[9];
        cs += P.dw_w[ky * 3 + kx] * (p[1] - Rex - Cex + cor);
      }
    }
    sPsum[t] = p[0] + cs;                      // sum(sigmoid) + sum(conv)
  }
  __syncthreads();
  // assemble[b,c] = pred_sum[b] * feat_sum[b,c]; stage head
  for (int i = t; i < 1024; i += 256) {
    sAsm[i] *= sPsum[i >> 6];
    sHead[i] = P.head[i];
  }
  __syncthreads();

  // ---- GEMMs (wave 0, EXEC all-ones) ----
  if (t < 32) {
    wmma_gemm(sAsm,  64, P.pt_w, P.pt_b, sPF, 128, 64, 128);  // pred_feat 16x128
    wmma_gemm(sHead, 64, P.ht_w, P.ht_b, sHF, 128, 64, 128);  // head_feat 16x128
  }
  __syncthreads();
  for (int i = t; i < 1024; i += 256) {
    int m = i >> 6, n = i & 63;
    sGate[i] = sHF[m * 128 + n] * sPF[m * 128 + n];           // gate_feat
  }
  __syncthreads();
  if (t < 32) {
    wmma_gemm(sGate, 64, P.hg_w, P.hg_b, sHG, 64, 64, 64);
    wmma_gemm(sGate, 64, P.pg_w, P.pg_b, sPG, 64, 64, 64);
  }
  __syncthreads();
  // LN stats: 0=HG(hni) 1=PG(pni) 2=head_out(hno) 3=pred_out(pno)
  if (t < 64) {
    int which = t >> 4, row = t & 15;
    const float* base =
        (which == 0) ? (sHG + row * 64) :
        (which == 1) ? (sPG + row * 64) :
        (which == 2) ? (sHF + row * 128 + 64) :
                       (sPF + row * 128 + 64);
    float mu, rs; ln_stats(base, &mu, &rs);
    sMu[which * 16 + row] = mu; sRs[which * 16 + row] = rs;
  }
  __syncthreads();
  for (int i = t; i < 1024; i += 256) {
    int m = i >> 6, n = i & 63;
    float hg = sigmoidf_((sHG[i] - sMu[m])      * sRs[m]      * P.hni_g[n] + P.hni_b[n]);
    float pg = sigmoidf_((sPG[i] - sMu[16 + m]) * sRs[16 + m] * P.pni_g[n] + P.pni_b[n]);
    float ho = (sHF[m * 128 + 64 + n] - sMu[32 + m]) * sRs[32 + m] * P.hno_g[n] + P.hno_b[n];
    float po = (sPF[m * 128 + 64 + n] - sMu[48 + m]) * sRs[48 + m] * P.pno_g[n] + P.pno_b[n];
    sUpd[i] = pg * po + hg * ho;
  }
  __syncthreads();
  if (t < 32) wmma_gemm(sUpd, 64, P.fc_w, P.fc_b, sU2, 64, 64, 64);
  __syncthreads();
  if (t < 16) {
    float mu, rs; ln_stats(sU2 + t * 64, &mu, &rs);
    sMu[64 + t] = mu; sRs[64 + t] = rs;
  }
  __syncthreads();
  for (int i = t; i < 1024; i += 256) {
    int m = i >> 6, n = i & 63;
    float v = (sU2[i] - sMu[64 + m]) * sRs[64 + m] * P.fcn_g[n] + P.fcn_b[n];
    P.out[i] = v > 0.f ? v : 0.f;                   // relu; out[b*64+c]
  }
}

extern "C" void kernel_launch(void* const* d_in, const int* in_sizes, int n_in,
                              void* d_out, int out_size, void* d_ws, size_t ws_size,
                              hipStream_t stream) {
  const float* feat = (const float*)d_in[0];
  const float* head = (const float*)d_in[1];
  const float* pred = (const float*)d_in[2];
  float* wsF = (float*)d_ws;          // 1024 * 32 floats
  float* wsP = wsF + 1024 * 32;       // 16 * 16 floats

  feat_reduce<<<1024, 256, 0, stream>>>(feat, wsF);
  pred_reduce<<<16, 256, 0, stream>>>(pred, wsP);

  Params P;
  P.wsF = wsF; P.wsP = wsP; P.head = head;
  P.dw_w  = (const float*)d_in[3];  P.dw_b  = (const float*)d_in[4];
  P.hw_w  = (const float*)d_in[5];  P.hw_b  = (const float*)d_in[6];
  P.ww_w  = (const float*)d_in[7];  P.ww_b  = (const float*)d_in[8];
  P.hh_w  = (const float*)d_in[9];  P.hh_b  = (const float*)d_in[10];
  P.pt_w  = (const float*)d_in[11]; P.pt_b  = (const float*)d_in[12];
  P.ht_w  = (const float*)d_in[13]; P.ht_b  = (const float*)d_in[14];
  P.pg_w  = (const float*)d_in[15]; P.pg_b  = (const float*)d_in[16];
  P.hg_w  = (const float*)d_in[17]; P.hg_b  = (const float*)d_in[18];
  P.fc_w  = (const float*)d_in[19]; P.fc_b  = (const float*)d_in[20];
  P.pni_g = (const float*)d_in[21]; P.pni_b = (const float*)d_in[22];
  P.hni_g = (const float*)d_in[23]; P.hni_b = (const float*)d_in[24];
  P.pno_g = (const float*)d_in[25]; P.pno_b = (const float*)d_in[26];
  P.hno_g = (const float*)d_in[27]; P.hno_b = (const float*)d_in[28];
  P.fcn_g = (const float*)d_in[29]; P.fcn_b = (const float*)d_in[30];
  P.out = (float*)d_out;

  head_finalize<<<1, 256, 0, stream>>>(P);
}